// OutputDecoder_3908420239676
// MI455X (gfx1250) — compile-verified
//
#include <hip/hip_runtime.h>
#include <stdint.h>

#define GRIDN 7
#define CELLS 49          // 7*7
#define NB    98          // 2*49 boxes
#define NC    20          // classes
#define MAXT  100
#define BLK   640         // 20 waves * 32 lanes (wave32), one wave per class
#define PREDF (CELLS*30)  // 1470 floats per image
#define PREDV2 (PREDF/2)  // 735 8-byte chunks (5880 bytes, always 8B-aligned)

__device__ __forceinline__ unsigned ballot32(bool p) {
  return __builtin_amdgcn_ballot_w32(p);
}

__global__ __launch_bounds__(BLK)
void yolo_decode_nms_kernel(const float* __restrict__ pred,
                            float* __restrict__ out, int B)
{
  const int b    = blockIdx.x;
  const int tid  = threadIdx.x;
  const int lane = tid & 31;
  const int wave = tid >> 5;          // wave == class id (0..19)

  // Output layout (flat, reference return order):
  //   num_val  : int32[B]
  //   boxes    : f32[B][100][4]
  //   scores   : f32[B][100]
  //   classes  : f32[B][100]
  int*   numv    = (int*)out;
  float* boxesO  = out + (size_t)B;
  float* scoresO = boxesO + (size_t)B * MAXT * 4;
  float* classO  = scoresO + (size_t)B * MAXT;

  __shared__ __align__(16) float s_pred[PREDF];  // raw prediction tile
  __shared__ float s_cmax[CELLS];        // per-cell max of relu'd class scores
  __shared__ float s_box[NB][4];         // decoded (y1,x1,y2,x2)
  __shared__ float s_scores[NC][NB];     // class-major scores
  __shared__ float s_sortScore[NC][NB];  // per-class sorted scores (desc)
  __shared__ int   s_sortIdx[NC][NB];    // per-class sorted box ids
  __shared__ float s_keptScore[NC][NB];  // NMS-kept, in rank order
  __shared__ int   s_keptIdx[NC][NB];
  __shared__ int   s_keptCnt[NC];

  // ------- Stage 0: async copy pred[b] into LDS (ASYNCcnt path, B64) -------
  // 5880 bytes/image: 8B-aligned for every b (5880 % 8 == 0), so b64 is the
  // widest safe async width (b128 would be misaligned on odd b).
  {
    const float* gp = pred + (size_t)b * PREDF;
    const unsigned ldsBase = (unsigned)(uintptr_t)(&s_pred[0]);
    for (int v = tid; v < PREDV2; v += BLK) {        // <=2 issues per lane
      unsigned ldsOff = ldsBase + (unsigned)(v * 8);
      unsigned long long ga = (unsigned long long)(const void*)(gp + v * 2);
      asm volatile("global_load_async_to_lds_b64 %0, %1, off"
                   :: "v"(ldsOff), "v"(ga) : "memory");
    }
    asm volatile("s_wait_asynccnt 0" ::: "memory");
  }
  __syncthreads();

  // ---------------- Stage 1: per-cell class max + box decode ---------------
  for (int q = tid; q < CELLS; q += BLK) {
    const float* pc = &s_pred[q * 30];
    float mm = -1.0f;
    #pragma unroll
    for (int c = 0; c < NC; ++c) mm = fmaxf(mm, fmaxf(pc[10 + c], 0.0f));
    s_cmax[q] = mm;
  }
  for (int j = tid; j < NB; j += BLK) {
    const int q = (j < CELLS) ? j : j - CELLS;
    const int o = (j < CELLS) ? 0 : 5;
    const float* pc = &s_pred[q * 30 + o];
    float p0 = fmaxf(pc[0], 0.0f), p1 = fmaxf(pc[1], 0.0f);
    float p2 = fmaxf(pc[2], 0.0f), p3 = fmaxf(pc[3], 0.0f);
    float cx = ((float)(q / GRIDN) + p0) * 64.0f;   // x_cells = first grid dim
    float cy = ((float)(q % GRIDN) + p1) * 64.0f;   // y_cells = second grid dim
    float w  = fminf(p2 * 448.0f, 448.0f);
    float h  = fminf(p3 * 448.0f, 448.0f);
    s_box[j][0] = cy - 0.5f * h;   // y1
    s_box[j][1] = cx - 0.5f * w;   // x1
    s_box[j][2] = cy + 0.5f * h;   // y2
    s_box[j][3] = cx + 0.5f * w;   // x2
  }
  __syncthreads();

  // ---------------- Stage 2: scores (conf * argmax-masked class) -----------
  for (int idx = tid; idx < NB * NC; idx += BLK) {
    const int j = idx / NC;
    const int c = idx - j * NC;
    const int q = (j < CELLS) ? j : j - CELLS;
    float conf = fmaxf(s_pred[q * 30 + ((j < CELLS) ? 4 : 9)], 0.0f);
    float clsv = fmaxf(s_pred[q * 30 + 10 + c], 0.0f);
    s_scores[c][j] = (clsv == s_cmax[q]) ? conf * clsv : 0.0f;
  }
  __syncthreads();

  // ---------------- Stage 3: per-class stable descending sort (rank) -------
  const int c = wave;
  #pragma unroll
  for (int t = 0; t < 4; ++t) {
    int j = lane + 32 * t;
    if (j < NB) {
      float sj = s_scores[c][j];
      int r = 0;
      for (int k = 0; k < NB; ++k) {
        float sk = s_scores[c][k];
        r += (sk > sj) || (sk == sj && k < j);   // stable: ties by lower index
      }
      s_sortScore[c][r] = sj;
      s_sortIdx[c][r]   = j;
    }
  }
  __syncthreads();

  // ---------------- Stage 4: greedy NMS, one wave per class ----------------
  {
    float bx0[4], bx1[4], bx2[4], bx3[4], area[4];
    bool  havej[4];
    #pragma unroll
    for (int t = 0; t < 4; ++t) {
      int j = lane + 32 * t;
      havej[t] = (j < NB);
      bx0[t] = bx1[t] = bx2[t] = bx3[t] = area[t] = 0.0f;
      if (havej[t]) {
        int bi = s_sortIdx[c][j];
        bx0[t] = s_box[bi][0]; bx1[t] = s_box[bi][1];
        bx2[t] = s_box[bi][2]; bx3[t] = s_box[bi][3];
        area[t] = (bx2[t] - bx0[t]) * (bx3[t] - bx1[t]);
      }
    }
    unsigned supBits = 0;   // bit t => sorted slot (lane + 32t) suppressed
    int kc = 0;
    for (int i = 0; i < NB; ++i) {
      float si = s_sortScore[c][i];        // uniform (broadcast LDS read)
      if (!(si > 0.05f)) break;            // sorted: nothing valid after this
      int ti = i >> 5, li = i & 31;
      unsigned m = ballot32((supBits >> ti) & 1u);
      if ((m >> li) & 1u) continue;        // slot i already suppressed
      int bi = s_sortIdx[c][i];
      if (lane == 0) { s_keptScore[c][kc] = si; s_keptIdx[c][kc] = bi; }
      kc++;
      float a0 = s_box[bi][0], a1 = s_box[bi][1];
      float a2 = s_box[bi][2], a3 = s_box[bi][3];
      float areaI = (a2 - a0) * (a3 - a1);
      #pragma unroll
      for (int t = 0; t < 4; ++t) {
        int j = lane + 32 * t;
        if (havej[t] && j > i && !((supBits >> t) & 1u)) {
          float iy = fmaxf(fminf(a2, bx2[t]) - fmaxf(a0, bx0[t]), 0.0f);
          float ix = fmaxf(fminf(a3, bx3[t]) - fmaxf(a1, bx1[t]), 0.0f);
          float inter = iy * ix;
          float uni   = areaI + area[t] - inter;
          float iou   = (uni > 0.0f) ? inter / uni : 0.0f;
          if (iou > 0.4f) supBits |= (1u << t);
        }
      }
    }
    if (lane == 0) s_keptCnt[c] = kc;
  }
  __syncthreads();

  // ---------------- Stage 5: 20-way merge -> global top-100 ----------------
  // Tie order matches top_k over flat index c*N + rank: lower class wins ties,
  // within a class rank order is preserved by popping front of the kept list.
  if (wave == 0) {
    int ptr = 0;
    int cnt = (lane < NC) ? s_keptCnt[lane] : 0;
    int nv  = 0;
    for (int t = 0; t < MAXT; ++t) {
      float hs = (lane < NC && ptr < cnt) ? s_keptScore[lane][ptr] : -1.0f;
      float bs = hs; int bl = lane;
      #pragma unroll
      for (int off = 16; off > 0; off >>= 1) {
        float os = __shfl_xor(bs, off, 32);
        int   ol = __shfl_xor(bl, off, 32);
        if (os > bs || (os == bs && ol < bl)) { bs = os; bl = ol; }
      }
      float* bo = boxesO + ((size_t)b * MAXT + t) * 4;
      if (bs > 0.0f) {
        int wptr = __shfl(ptr, bl, 32);
        if (lane == 0) {
          int bi = s_keptIdx[bl][wptr];
          bo[0] = s_box[bi][0]; bo[1] = s_box[bi][1];
          bo[2] = s_box[bi][2]; bo[3] = s_box[bi][3];
          scoresO[(size_t)b * MAXT + t] = bs;
          classO [(size_t)b * MAXT + t] = (float)bl;
        }
        if (lane == bl) ptr++;
        nv++;
      } else {
        if (lane == 0) {
          bo[0] = bo[1] = bo[2] = bo[3] = 0.0f;
          scoresO[(size_t)b * MAXT + t] = 0.0f;
          classO [(size_t)b * MAXT + t] = 0.0f;
        }
      }
    }
    if (lane == 0) numv[b] = nv;   // int32 count in output slot 0 region
  }
}

extern "C" void kernel_launch(void* const* d_in, const int* in_sizes, int n_in,
                              void* d_out, int out_size, void* d_ws, size_t ws_size,
                              hipStream_t stream) {
  (void)n_in; (void)out_size; (void)d_ws; (void)ws_size;
  const float* pred = (const float*)d_in[0];
  float* out = (float*)d_out;
  const int B = in_sizes[0] / PREDF;     // 4096
  hipLaunchKernelGGL(yolo_decode_nms_kernel, dim3(B), dim3(BLK), 0, stream,
                     pred, out, B);
}